// my_MetaPath2Vec_16724602650996
// MI455X (gfx1250) — compile-verified
//
#include <hip/hip_runtime.h>

typedef __attribute__((ext_vector_type(2))) float v2f;
typedef __attribute__((ext_vector_type(8))) float v8f;

#define BLOCK 256
#define WAVES_PER_BLOCK 8
#define GRID_A 4096
#define EPS_F 1e-15f

// DPP8 lane-select immediate: sel[i] chooses the source lane within each
// aligned 8-lane group.
#define DPP8_SEL(s0, s1, s2, s3, s4, s5, s6, s7)                         \
    ((s0) | ((s1) << 3) | ((s2) << 6) | ((s3) << 9) | ((s4) << 12) |     \
     ((s5) << 15) | ((s6) << 18) | ((s7) << 21))

#define DPP8_XOR4 DPP8_SEL(4, 5, 6, 7, 0, 1, 2, 3)
#define DPP8_XOR2 DPP8_SEL(2, 3, 0, 1, 6, 7, 4, 5)
#define DPP8_XOR1 DPP8_SEL(1, 0, 3, 2, 5, 4, 7, 6)

// Butterfly exchange inside an 8-lane group on the VALU (v_mov_dpp8) instead
// of the LDS crossbar (ds_bpermute) -> zero DS traffic in the hot loop.
template <int IMM>
__device__ __forceinline__ float xadd_dpp8(float v) {
    int j = __builtin_amdgcn_mov_dpp8(__builtin_bit_cast(int, v), IMM);
    return v + __builtin_bit_cast(float, j);
}

// Exact fp32 block reduction (256 values) using V_WMMA_F32_16X16X4_F32 with a
// ones B-matrix: D[i][j] = sum_k A[i][k] * 1.0 + C[i][j].  Chaining 4 WMMAs
// over the 256 staged values makes every D row a partial total; summing the 8
// D components per lane and adding the two half-wave partials yields the exact
// total.  Because B is all-ones and only the grand total is consumed, the
// result is independent of the A/B VGPR layouts -> layout-robust, exact fp32.
// Valid result returned in wave 0 (others get 0).
__device__ __forceinline__ float block_reduce_256(float v, float* lds) {
    lds[threadIdx.x] = v;
    __syncthreads();
    float total = 0.0f;
    if (threadIdx.x < 32) {           // wave 0 fully active -> EXEC all ones
        const int lane = threadIdx.x;
        const v2f ones = {1.0f, 1.0f};
        v8f d = {0.f, 0.f, 0.f, 0.f, 0.f, 0.f, 0.f, 0.f};
#pragma unroll
        for (int t = 0; t < 4; ++t) {
            v2f a;
            a.x = lds[t * 64 + lane * 2 + 0];
            a.y = lds[t * 64 + lane * 2 + 1];
            d = __builtin_amdgcn_wmma_f32_16x16x4_f32(
                    false, a, false, ones, (short)0, d, false, false);
        }
        float s = d[0] + d[1] + d[2] + d[3] + d[4] + d[5] + d[6] + d[7];
        total = s + __shfl_xor(s, 16, 32);   // rows 0-7 half + rows 8-15 half
    }
    __syncthreads();                  // allow caller to reuse lds
    return total;
}

// 4 walks per wave32, 8 lanes per walk.  Each lane owns 16 consecutive floats
// of a 128-float row (4 x float4); per-lane dot partials cover 16 elements and
// the cross-lane reduction is 3 DPP8 butterfly rounds on the VALU.  Every
// global_load_b128 moves 512B: per walk, eight 16B pieces at a 64B stride.
// sgn=+1: log(sigmoid(s)+eps); sgn=-1 uses the exact 1-sigmoid(s)=sigmoid(-s).
__global__ __launch_bounds__(BLOCK)
void walk_loss_kernel(const float* __restrict__ emb,
                      const long long* __restrict__ rw,
                      long long n_walks, float sgn,
                      float* __restrict__ partials) {
    __shared__ float lds[BLOCK];
    const int lane = threadIdx.x & 31;
    const int sub  = lane & 7;          // position within 8-lane group
    const int slot = lane >> 3;         // which of the 4 walks in this wave

    const long long wave    = (long long)blockIdx.x * WAVES_PER_BLOCK + (threadIdx.x >> 5);
    const long long nwaves  = (long long)gridDim.x * WAVES_PER_BLOCK;
    const long long ngroups = (n_walks + 3) >> 2;

    float acc = 0.0f;
    for (long long g = wave; g < ngroups; g += nwaves) {
        const long long wid   = g * 4 + slot;
        const bool      valid = (wid < n_walks);
        const long long wsafe = valid ? wid : (n_walks - 1);   // keep loads in-bounds
        const long long* r    = rw + wsafe * 7;

        if (g + nwaves < ngroups)                     // gfx1250 global_prefetch_b8
            __builtin_prefetch(rw + ((g + nwaves) * 4 + slot) * 7, 0, 1);

        // Start row: 16 resident floats per lane.
        const long long i0 = r[0];
        const float4* s4 = (const float4*)(emb + i0 * 128) + sub * 4;
        const float4 s0 = s4[0], s1 = s4[1], s2 = s4[2], s3 = s4[3];

        // Stream the 6 context rows: load 64B, FMA into p[c], recycle regs.
        float p[6];
#pragma unroll
        for (int c = 0; c < 6; ++c) {
            const long long ix = r[c + 1];
            const float4* c4 = (const float4*)(emb + ix * 128) + sub * 4;
            const float4 c0 = c4[0], c1 = c4[1], c2 = c4[2], c3 = c4[3];
            float t0 = s0.x * c0.x + s0.y * c0.y + s0.z * c0.z + s0.w * c0.w;
            float t1 = s1.x * c1.x + s1.y * c1.y + s1.z * c1.z + s1.w * c1.w;
            float t2 = s2.x * c2.x + s2.y * c2.y + s2.z * c2.z + s2.w * c2.w;
            float t3 = s3.x * c3.x + s3.y * c3.y + s3.z * c3.z + s3.w * c3.w;
            p[c] = (t0 + t1) + (t2 + t3);
        }

        // VALU butterfly over the 8-lane group (xor 4, 2, 1 via DPP8).
#pragma unroll
        for (int c = 0; c < 6; ++c) p[c] = xadd_dpp8<DPP8_XOR4>(p[c]);
#pragma unroll
        for (int c = 0; c < 6; ++c) p[c] = xadd_dpp8<DPP8_XOR2>(p[c]);
#pragma unroll
        for (int c = 0; c < 6; ++c) p[c] = xadd_dpp8<DPP8_XOR1>(p[c]);

        // Lanes sub=0..5 of each group evaluate one log-sigmoid term.
        if (valid && sub < 6) {
            float ps = p[0];
#pragma unroll
            for (int c = 1; c < 6; ++c) ps = (sub == c) ? p[c] : ps;
            const float x  = ps * sgn;
            const float sg = 1.0f / (1.0f + __expf(-x));
            acc += __logf(sg + EPS_F);
        }
    }

    const float tot = block_reduce_256(acc, lds);
    if (threadIdx.x == 0) partials[blockIdx.x] = tot;   // every block writes
}

__global__ __launch_bounds__(BLOCK)
void final_reduce_kernel(const float* __restrict__ posp,
                         const float* __restrict__ negp,
                         int n, float inv_pos, float inv_neg,
                         float* __restrict__ out) {
    __shared__ float lds[BLOCK];
    float sp = 0.0f, sn = 0.0f;
    for (int i = threadIdx.x; i < n; i += BLOCK) {      // fixed order -> deterministic
        sp += posp[i];
        sn += negp[i];
    }
    const float SP = block_reduce_256(sp, lds);
    const float SN = block_reduce_256(sn, lds);
    if (threadIdx.x == 0)
        out[0] = -(SP * inv_pos) - (SN * inv_neg);
}

extern "C" void kernel_launch(void* const* d_in, const int* in_sizes, int n_in,
                              void* d_out, int out_size, void* d_ws, size_t ws_size,
                              hipStream_t stream) {
    const float*     emb    = (const float*)d_in[0];
    const long long* pos_rw = (const long long*)d_in[1];   // int64 walks
    const long long* neg_rw = (const long long*)d_in[2];

    const long long n_pos = (long long)in_sizes[1] / 7;
    const long long n_neg = (long long)in_sizes[2] / 7;

    float* posp = (float*)d_ws;            // GRID_A floats
    float* negp = posp + GRID_A;           // GRID_A floats  (32 KB total scratch)

    walk_loss_kernel<<<GRID_A, BLOCK, 0, stream>>>(emb, pos_rw, n_pos,  1.0f, posp);
    walk_loss_kernel<<<GRID_A, BLOCK, 0, stream>>>(emb, neg_rw, n_neg, -1.0f, negp);

    const float inv_pos = 1.0f / (float)(n_pos * 6);
    const float inv_neg = 1.0f / (float)(n_neg * 6);
    final_reduce_kernel<<<1, BLOCK, 0, stream>>>(posp, negp, GRID_A,
                                                 inv_pos, inv_neg, (float*)d_out);
}